// GCNPair_37074157699766
// MI455X (gfx1250) — compile-verified
//
#include <hip/hip_runtime.h>

// ---------------------------------------------------------------------------
// GCN pair model on MI455X (gfx1250, wave32).
// Dense GEMMs use V_WMMA_F32_16X16X4_F32 (full fp32, matching the reference).
// Sparse scatter uses wave-per-edge float4 gathers + global_atomic_add_f32
// (L2-resident: node features are 25.6 MB << 192 MB L2).
// ---------------------------------------------------------------------------

typedef __attribute__((ext_vector_type(2))) float v2f;
typedef __attribute__((ext_vector_type(8))) float v8f;

#define NODE_F 128  // feature width of all conv GEMM outputs

// ---------------------------------------------------------------------------
// GEMM: C[rows x 128] = A[rows x K] * W[K x 128] (+bias, +relu optional)
// Block = 256 threads = 8 waves. Block computes a 16-row x 128-col tile;
// wave w computes cols [16w, 16w+16). W and the 16xK A-tile live in LDS.
// WMMA f32 16x16x4 layouts (ISA 7.12.2):
//   A (16x4):  VGPR0 = {lanes 0-15: K=k,   lanes16-31: K=k+2}
//              VGPR1 = {lanes 0-15: K=k+1, lanes16-31: K=k+3}
//   B (4x16):  mirrored (row striped across lanes within a VGPR)
//   C/D 16x16: VGPR r -> M = r + 8*(lane>=16), N = lane&15
// ---------------------------------------------------------------------------
__global__ void gemm_wmma_f32(const float* __restrict__ A,
                              const float* __restrict__ W,
                              const float* __restrict__ bias,
                              float* __restrict__ C,
                              int rows, int K, int relu) {
  extern __shared__ float lds[];
  float* Ws = lds;                 // K * 128
  float* Xs = lds + (size_t)K * NODE_F;  // 16 * (K+1)  (padded vs bank conflicts)
  const int KS = K + 1;

  const int row0 = blockIdx.x * 16;
  const int tid  = threadIdx.x;

  // Stage W (K x 128) into LDS
  for (int i = tid; i < K * NODE_F; i += 256) Ws[i] = W[i];
  // Stage 16 x K tile of A into LDS (zero-pad past `rows`)
  for (int i = tid; i < 16 * K; i += 256) {
    int r = i / K, c = i % K;
    int gr = row0 + r;
    Xs[r * KS + c] = (gr < rows) ? A[(size_t)gr * K + c] : 0.0f;
  }
  __syncthreads();

  const int wave = tid >> 5;       // 8 waves (wave32)
  const int lane = tid & 31;
  const int col0 = wave * 16;
  const int half = lane >> 4;      // 0: lanes 0-15, 1: lanes 16-31
  const int lidx = lane & 15;

  v8f acc = {};
  for (int k = 0; k < K; k += 4) {
    const int ka = k + half * 2;
    v2f a, b;
    a.x = Xs[lidx * KS + ka];
    a.y = Xs[lidx * KS + ka + 1];
    b.x = Ws[(size_t)ka * NODE_F + col0 + lidx];
    b.y = Ws[(size_t)(ka + 1) * NODE_F + col0 + lidx];
    acc = __builtin_amdgcn_wmma_f32_16x16x4_f32(
        /*neg_a=*/false, a, /*neg_b=*/false, b,
        /*c_mod=*/(short)0, acc, /*reuse_a=*/false, /*reuse_b=*/false);
  }

  // Write back 16x16 tile (C/D VGPR layout)
  for (int r = 0; r < 8; ++r) {
    const int row = r + half * 8;
    const int gr  = row0 + row;
    if (gr < rows) {
      const int gc = col0 + lidx;
      float v = acc[r];
      if (bias) v += bias[gc];
      if (relu) v = fmaxf(v, 0.0f);
      C[(size_t)gr * NODE_F + gc] = v;
    }
  }
}

// ---------------------------------------------------------------------------
// deg[dst] += 1 per edge (self-loop +1 folded into dinv kernel)
// ---------------------------------------------------------------------------
__global__ void deg_kernel(const int* __restrict__ ei, float* __restrict__ deg, int E) {
  int e = blockIdx.x * blockDim.x + threadIdx.x;
  if (e < E) unsafeAtomicAdd(&deg[ei[E + e]], 1.0f);
}

__global__ void dinv_kernel(float* __restrict__ deg, int n) {
  int i = blockIdx.x * blockDim.x + threadIdx.x;
  if (i < n) deg[i] = rsqrtf(deg[i] + 1.0f);  // deg+1 >= 1 always
}

// out[v,c] = h[v,c] * dinv[v]^2 + bias[c]   (self-loop term + bias, init acc)
__global__ void init_out_kernel(const float* __restrict__ h,
                                const float* __restrict__ dinv,
                                const float* __restrict__ bias,
                                float* __restrict__ out, int n) {
  int i = blockIdx.x * blockDim.x + threadIdx.x;
  if (i >= n * NODE_F) return;
  int v = i >> 7, c = i & (NODE_F - 1);
  float d = dinv[v];
  out[i] = h[i] * d * d + bias[c];
}

// One wave32 per edge: out[dst,:] += h[src,:] * dinv[src]*dinv[dst]
// Each lane moves 4 consecutive floats (32*4 = 128).
__global__ void scatter_kernel(const float* __restrict__ h,
                               const int* __restrict__ ei,
                               const float* __restrict__ dinv,
                               float* __restrict__ out, int E) {
  int gt   = blockIdx.x * blockDim.x + threadIdx.x;
  int e    = gt >> 5;
  int lane = gt & 31;
  if (e >= E) return;
  int src = ei[e];
  int dst = ei[E + e];
  float norm = dinv[src] * dinv[dst];
  const float4* hp = (const float4*)(h + (size_t)src * NODE_F);
  float4 v = hp[lane];
  float* op = out + (size_t)dst * NODE_F + lane * 4;
  unsafeAtomicAdd(op + 0, v.x * norm);
  unsafeAtomicAdd(op + 1, v.y * norm);
  unsafeAtomicAdd(op + 2, v.z * norm);
  unsafeAtomicAdd(op + 3, v.w * norm);
}

__global__ void relu_kernel(float* __restrict__ x, int n) {
  int i = blockIdx.x * blockDim.x + threadIdx.x;
  if (i < n) x[i] = fmaxf(x[i], 0.0f);
}

// merged[p, 0:128] = h_l[labels[p,0]]; merged[p, 128:256] = h_r[labels[p,1]]
__global__ void gather_kernel(const float* __restrict__ hl,
                              const float* __restrict__ hr,
                              const int* __restrict__ labels,
                              float* __restrict__ merged, int P) {
  int i = blockIdx.x * blockDim.x + threadIdx.x;
  if (i >= P * 256) return;
  int p = i >> 8, c = i & 255;
  int node = (c < 128) ? labels[p * 3 + 0] : labels[p * 3 + 1];
  const float* src = (c < 128) ? hl : hr;
  merged[i] = src[(size_t)node * NODE_F + (c & 127)];
}

// out[p, o] = sum_k h[p,k] * Wfc2[k,o] + b[o]   (O=2: VALU dot, negligible)
__global__ void fc2_kernel(const float* __restrict__ h,
                           const float* __restrict__ W,
                           const float* __restrict__ b,
                           float* __restrict__ out, int P, int O) {
  int i = blockIdx.x * blockDim.x + threadIdx.x;
  if (i >= P * O) return;
  int p = i / O, o = i % O;
  float acc = b[o];
  const float* hp = h + (size_t)p * NODE_F;
#pragma unroll 4
  for (int k = 0; k < NODE_F; ++k) acc += hp[k] * W[k * O + o];
  out[i] = acc;
}

// ---------------------------------------------------------------------------
extern "C" void kernel_launch(void* const* d_in, const int* in_sizes, int n_in,
                              void* d_out, int out_size, void* d_ws, size_t ws_size,
                              hipStream_t stream) {
  const float* x_l  = (const float*)d_in[0];
  const int*   ei_l = (const int*)  d_in[1];
  const float* x_r  = (const float*)d_in[2];
  const int*   ei_r = (const int*)  d_in[3];
  const int*   lab  = (const int*)  d_in[4];
  const float* W1_l = (const float*)d_in[5];
  const float* b1_l = (const float*)d_in[6];
  const float* W2_l = (const float*)d_in[7];
  const float* b2_l = (const float*)d_in[8];
  const float* W1_r = (const float*)d_in[9];
  const float* b1_r = (const float*)d_in[10];
  const float* W2_r = (const float*)d_in[11];
  const float* b2_r = (const float*)d_in[12];
  const float* Wfc1 = (const float*)d_in[13];
  const float* bfc1 = (const float*)d_in[14];
  const float* Wfc2 = (const float*)d_in[15];
  const float* bfc2 = (const float*)d_in[16];
  float* out = (float*)d_out;

  const int n = in_sizes[0] / NODE_F;  // 50000
  const int E = in_sizes[1] / 2;       // 600000
  const int P = in_sizes[4] / 3;       // 4096
  const int O = in_sizes[16];          // 2 (bfc2 length)

  // Workspace layout (floats)
  const size_t NB = (size_t)n * NODE_F;
  float* bufA   = (float*)d_ws;        // gemm scratch
  float* bufB   = bufA + NB;           // h1 per branch
  float* h2l    = bufB + NB;
  float* h2r    = h2l + NB;
  float* deg_l  = h2r + NB;            // becomes dinv in place
  float* deg_r  = deg_l + n;
  float* merged = deg_r + n;           // P x 256
  float* fc1o   = merged + (size_t)P * 256;  // P x 128

  const int BT = 256;
  const dim3 blk(BT);

  // ---- degrees / dinv (both graphs) ----
  hipMemsetAsync(deg_l, 0, 2 * (size_t)n * sizeof(float), stream);
  deg_kernel<<<(E + BT - 1) / BT, blk, 0, stream>>>(ei_l, deg_l, E);
  deg_kernel<<<(E + BT - 1) / BT, blk, 0, stream>>>(ei_r, deg_r, E);
  dinv_kernel<<<(n + BT - 1) / BT, blk, 0, stream>>>(deg_l, n);
  dinv_kernel<<<(n + BT - 1) / BT, blk, 0, stream>>>(deg_r, n);

  const int gemm_blocks = (n + 15) / 16;
  const size_t lds128 = (size_t)(128 * NODE_F + 16 * 129) * sizeof(float);  // ~74 KB
  const size_t lds256 = (size_t)(256 * NODE_F + 16 * 257) * sizeof(float);  // ~148 KB
  const int nfThreads = n * NODE_F;
  const int nfBlocks  = (nfThreads + BT - 1) / BT;
  const int edgeBlocks = (E * 32 + BT - 1) / BT;  // one wave32 per edge

  // ---- per-branch GCN: conv1(relu) -> conv2 ----
  auto branch = [&](const float* x, const int* ei, const float* dinv,
                    const float* W1, const float* b1,
                    const float* W2, const float* b2, float* h2) {
    // conv1: h = x@W1 ; acc = selfloop+bias ; scatter ; relu
    gemm_wmma_f32<<<gemm_blocks, blk, lds128, stream>>>(x, W1, nullptr, bufA, n, 128, 0);
    init_out_kernel<<<nfBlocks, blk, 0, stream>>>(bufA, dinv, b1, bufB, n);
    scatter_kernel<<<edgeBlocks, blk, 0, stream>>>(bufA, ei, dinv, bufB, E);
    relu_kernel<<<nfBlocks, blk, 0, stream>>>(bufB, nfThreads);
    // conv2
    gemm_wmma_f32<<<gemm_blocks, blk, lds128, stream>>>(bufB, W2, nullptr, bufA, n, 128, 0);
    init_out_kernel<<<nfBlocks, blk, 0, stream>>>(bufA, dinv, b2, h2, n);
    scatter_kernel<<<edgeBlocks, blk, 0, stream>>>(bufA, ei, dinv, h2, E);
  };
  branch(x_l, ei_l, deg_l, W1_l, b1_l, W2_l, b2_l, h2l);
  branch(x_r, ei_r, deg_r, W1_r, b1_r, W2_r, b2_r, h2r);

  // ---- pair head ----
  gather_kernel<<<((P * 256) + BT - 1) / BT, blk, 0, stream>>>(h2l, h2r, lab, merged, P);
  gemm_wmma_f32<<<(P + 15) / 16, blk, lds256, stream>>>(merged, Wfc1, bfc1, fc1o, P, 256, 1);
  fc2_kernel<<<((P * O) + BT - 1) / BT, blk, 0, stream>>>(fc1o, Wfc2, bfc2, out, P, O);
}